// NMS_1047972020518
// MI455X (gfx1250) — compile-verified
//
#include <hip/hip_runtime.h>
#include <hip/hip_bf16.h>
#include <stdint.h>

// ---------------------------------------------------------------------------
// YOLO-face NMS for MI455X (gfx1250, wave32)
//   B=64 images, N=25200 candidates, C=16 channels, TOP_K=1024
// Kernel 1 (per-image block): score -> 64-bit keys -> LDS bitonic sort
//           (32768 x u64 = 256KB LDS, fits the 320KB WGP LDS) -> top-1024
//           order/score/xyxy boxes into workspace.
// Kernel 2 (per-image block): async-to-LDS box staging, suppression bitmask
//           via WMMA f32 16x16x4 (area_i + area_j outer sum) + branchless
//           VALU intersection, sequential greedy reduce on wave 0, output.
// ---------------------------------------------------------------------------

#define BATCH      64
#define NCAND      25200
#define NCH        16
#define TOPK       1024
#define SORT_N     32768
#define CONF_T     0.25f
#define IOU_T      0.45f

typedef __attribute__((ext_vector_type(2))) float v2f;
typedef __attribute__((ext_vector_type(8))) float v8f;
typedef int v4i __attribute__((vector_size(16)));   // 128-bit payload type
typedef __attribute__((address_space(1))) v4i* g_v4i_ptr;   // global
typedef __attribute__((address_space(3))) v4i* l_v4i_ptr;   // LDS

// --------------------------------------------------------------------------
// Kernel 1: score + full bitonic sort in LDS + emit top-1024 order/boxes
// --------------------------------------------------------------------------
__global__ __launch_bounds__(1024, 1)
void nms_topk_kernel(const float* __restrict__ x,
                     int*   __restrict__ ordidx,   // [B][TOPK]
                     float* __restrict__ tops,     // [B][TOPK]
                     float* __restrict__ boxesW)   // [B][TOPK][4] xyxy
{
    extern __shared__ unsigned char dynsmem[];
    unsigned long long* keys = (unsigned long long*)dynsmem;   // SORT_N

    const int b   = blockIdx.x;
    const int tid = threadIdx.x;

    // Build sortable keys: (monotone(score) << 32) | ~idx  (desc score, asc idx)
    for (int i = tid; i < SORT_N; i += 1024) {
        unsigned long long k = 0ull;
        if (i < NCAND) {
            const float* p = x + ((size_t)b * NCAND + i) * NCH;
            float sc = p[4] * p[15];
            float s  = (sc > CONF_T) ? sc : -1.0f;
            unsigned ub = __float_as_uint(s);
            ub = (ub & 0x80000000u) ? ~ub : (ub | 0x80000000u);
            k = ((unsigned long long)ub << 32) | (unsigned)(~(unsigned)i);
        }
        keys[i] = k;
    }
    __syncthreads();

    // Bitonic sort, descending. 15*16/2 = 120 passes, 16 CE per thread each.
    for (unsigned ksz = 2; ksz <= SORT_N; ksz <<= 1) {
        for (unsigned j = ksz >> 1; j > 0; j >>= 1) {
            for (unsigned t = tid; t < (SORT_N / 2); t += 1024) {
                unsigned i   = ((t & ~(j - 1u)) << 1) | (t & (j - 1u));
                unsigned ixj = i | j;
                unsigned long long a = keys[i];
                unsigned long long c = keys[ixj];
                bool desc = ((i & ksz) == 0);
                if (desc ? (a < c) : (a > c)) { keys[i] = c; keys[ixj] = a; }
            }
            __syncthreads();
        }
    }

    // Emit top-1024: order index, score, xyxy box.
    {
        unsigned long long k = keys[tid];
        unsigned idx = ~(unsigned)k;
        unsigned ub  = (unsigned)(k >> 32);
        unsigned sb  = (ub & 0x80000000u) ? (ub & 0x7FFFFFFFu) : ~ub;
        float s = __uint_as_float(sb);
        if (idx >= (unsigned)NCAND) { idx = 0; s = -1.0f; }

        ordidx[b * TOPK + tid] = (int)idx;
        tops  [b * TOPK + tid] = s;

        const float* p = x + ((size_t)b * NCAND + idx) * NCH;
        float cx = p[0], cy = p[1], w = p[2], h = p[3];
        float* bx = boxesW + ((size_t)b * TOPK + tid) * 4;
        bx[0] = cx - 0.5f * w;
        bx[1] = cy - 0.5f * h;
        bx[2] = cx + 0.5f * w;
        bx[3] = cy + 0.5f * h;
    }
}

// --------------------------------------------------------------------------
// Kernel 2: WMMA-assisted IoU mask + greedy suppression + output
// --------------------------------------------------------------------------
__global__ __launch_bounds__(1024, 1)
void nms_suppress_kernel(const float* __restrict__ x,
                         const int*   __restrict__ ordidx,
                         const float* __restrict__ tops,
                         const float* __restrict__ boxesW,
                         float* __restrict__ out)   // [B][TOPK][16]
{
    extern __shared__ unsigned char dynsmem[];
    float*    sbox  = (float*)dynsmem;            // [TOPK*4]
    float*    sarea = sbox + TOPK * 4;            // [TOPK]
    float*    stops = sarea + TOPK;               // [TOPK]
    unsigned* smask = (unsigned*)(stops + TOPK);  // [TOPK*32] suppression bits
    unsigned* skeep = smask + TOPK * 32;          // [32]

    const int b    = blockIdx.x;
    const int tid  = threadIdx.x;
    const int lane = tid & 31;
    const int wave = tid >> 5;

    // Stage boxes into LDS: one async B128 per thread (ASYNCcnt-tracked DMA
    // path, bypasses VGPRs) when the toolchain exposes it.
#if __has_builtin(__builtin_amdgcn_global_load_async_to_lds_b128)
    {
        const float* gsrc = boxesW + (size_t)b * TOPK * 4 + tid * 4;
        float*       ldst = sbox + tid * 4;
        __builtin_amdgcn_global_load_async_to_lds_b128(
            (g_v4i_ptr)(gsrc),
            (l_v4i_ptr)(ldst),
            0, 0);
#if __has_builtin(__builtin_amdgcn_s_wait_asynccnt)
        __builtin_amdgcn_s_wait_asynccnt(0);
#else
        asm volatile("s_wait_asynccnt 0x0" ::: "memory");
#endif
    }
#else
    for (int i = tid; i < TOPK * 4; i += 1024)
        sbox[i] = boxesW[(size_t)b * TOPK * 4 + i];
#endif
    __syncthreads();
    {
        float x1 = sbox[tid * 4 + 0], y1 = sbox[tid * 4 + 1];
        float x2 = sbox[tid * 4 + 2], y2 = sbox[tid * 4 + 3];
        sarea[tid] = fmaxf(x2 - x1, 0.0f) * fmaxf(y2 - y1, 0.0f);
        stops[tid] = tops[b * TOPK + tid];
    }
    __syncthreads();

    // Build 1024x1024 suppression bitmask as [row][32 words].
    // Strips: 64 row-tiles (16 rows) x 32 col-strips (32 cols) = 2048, 64/wave.
    const float loMask = (lane < 16) ? 1.0f : 0.0f;   // branch-free lane select
    for (int s = wave; s < 64 * 32; s += 32) {
        const int rowBase = (s >> 5) * 16;
        const int colBase = (s & 31) * 32;

        // A fragment: 16x4 f32, rows carry [area_m, 1, 0, 0].
        // Unconditional LDS load + cndmask (no EXEC branches).
        float aRow = sarea[rowBase + (lane & 15)];
        v2f A;
        A.x = aRow * loMask;   // K=0 (lanes 0-15), K=2 = 0 (lanes 16-31)
        A.y = loMask;          // K=1 = 1.0,        K=3 = 0

        unsigned words[16];
        #pragma unroll
        for (int t = 0; t < 2; ++t) {
            const int cb   = colBase + t * 16;
            const int nloc = lane & 15;
            const int ncol = cb + nloc;

            // B fragment: 4x16 f32, K=0 row = ones, K=1 row = area_n.
            float aCol = sarea[cb + nloc];
            v2f Bf;
            Bf.x = loMask;
            Bf.y = aCol * loMask;
            v8f Cz = {};
            v8f D;
#if __has_builtin(__builtin_amdgcn_wmma_f32_16x16x4_f32)
            // D[m][n] = area_m + area_n (exact rank-2 outer sum via WMMA)
            D = __builtin_amdgcn_wmma_f32_16x16x4_f32(
                    false, A, false, Bf, (short)0, Cz, false, false);
#else
            #pragma unroll
            for (int r = 0; r < 8; ++r) {
                int mrow = rowBase + r + ((lane >= 16) ? 8 : 0);
                D[r] = sarea[mrow] + sarea[ncol];
            }
            (void)A; (void)Bf; (void)Cz;
#endif
            float cx1 = sbox[ncol * 4 + 0], cy1 = sbox[ncol * 4 + 1];
            float cx2 = sbox[ncol * 4 + 2], cy2 = sbox[ncol * 4 + 3];

            #pragma unroll
            for (int r = 0; r < 8; ++r) {
                int mrow = rowBase + r + ((lane >= 16) ? 8 : 0);
                float rx1 = sbox[mrow * 4 + 0], ry1 = sbox[mrow * 4 + 1];
                float rx2 = sbox[mrow * 4 + 2], ry2 = sbox[mrow * 4 + 3];
                float iw = fmaxf(fminf(rx2, cx2) - fmaxf(rx1, cx1), 0.0f);
                float ih = fmaxf(fminf(ry2, cy2) - fmaxf(ry1, cy1), 0.0f);
                float inter = iw * ih;
                float uni   = D[r] - inter;
                // Bitwise & (not &&): keep the predicate branch-free.
                int pred = (int)(ncol > mrow) & (int)(inter > IOU_T * (uni + 1e-9f));
                words[t * 8 + r] = (unsigned)__ballot(pred);
            }
        }
        if (lane == 0) {
            #pragma unroll
            for (int r = 0; r < 8; ++r) {
                unsigned b0 = words[r], b1 = words[8 + r];
                // ballot bits [15:0] = M=r rows, [31:16] = M=r+8 rows
                smask[(rowBase + r)     * 32 + (s & 31)] =
                    ((b1 & 0xFFFFu) << 16) | (b0 & 0xFFFFu);
                smask[(rowBase + r + 8) * 32 + (s & 31)] =
                    (b1 & 0xFFFF0000u) | (b0 >> 16);
            }
        }
    }
    __syncthreads();

    // Sequential greedy reduce on wave 0: lane owns one 32-bit keep word.
    if (tid < 32) {
        unsigned kw = 0;
        for (int bit = 0; bit < 32; ++bit)
            if (stops[tid * 32 + bit] > 0.0f) kw |= (1u << bit);
        for (int i = 0; i < TOPK; ++i) {
            int wi = i >> 5, bi = i & 31;
            unsigned row = (unsigned)__shfl((int)kw, wi, 32);
            if ((row >> bi) & 1u) kw &= ~smask[i * 32 + tid];
        }
        skeep[tid] = kw;
    }
    __syncthreads();

    // Output: [x1 y1 x2 y2 conf lm0..lm9 0] or zeros.
    {
        const int k   = tid;
        bool kept = (skeep[k >> 5] >> (k & 31)) & 1u;
        float* o = out + ((size_t)b * TOPK + k) * NCH;
        if (kept) {
            o[0] = sbox[k * 4 + 0];
            o[1] = sbox[k * 4 + 1];
            o[2] = sbox[k * 4 + 2];
            o[3] = sbox[k * 4 + 3];
            o[4] = stops[k];
            int oi = ordidx[b * TOPK + k];
            const float* p = x + ((size_t)b * NCAND + oi) * NCH;
            #pragma unroll
            for (int j = 0; j < 10; ++j) o[5 + j] = p[5 + j];
            o[15] = 0.0f;
        } else {
            #pragma unroll
            for (int j = 0; j < 16; ++j) o[j] = 0.0f;
        }
    }
}

// --------------------------------------------------------------------------
extern "C" void kernel_launch(void* const* d_in, const int* in_sizes, int n_in,
                              void* d_out, int out_size, void* d_ws, size_t ws_size,
                              hipStream_t stream) {
    (void)in_sizes; (void)n_in; (void)out_size; (void)ws_size;
    const float* x = (const float*)d_in[0];
    float* out = (float*)d_out;

    // Workspace layout: order idx | top scores | xyxy boxes
    int*   ordidx = (int*)d_ws;                       // B*TOPK
    float* tops   = (float*)(ordidx + BATCH * TOPK);  // B*TOPK
    float* boxesW = tops + BATCH * TOPK;              // B*TOPK*4

    const size_t smem1 = (size_t)SORT_N * sizeof(unsigned long long);      // 256 KB
    const size_t smem2 = (size_t)(TOPK * 4 + TOPK + TOPK) * sizeof(float)  // boxes+area+tops
                       + (size_t)(TOPK * 32 + 32) * sizeof(unsigned);      // mask+keep  ~152 KB

    // Idempotent, deterministic: raise dynamic-LDS cap every call (no-op after first).
    (void)hipFuncSetAttribute((const void*)nms_topk_kernel,
                              hipFuncAttributeMaxDynamicSharedMemorySize, (int)smem1);
    (void)hipFuncSetAttribute((const void*)nms_suppress_kernel,
                              hipFuncAttributeMaxDynamicSharedMemorySize, (int)smem2);

    nms_topk_kernel<<<BATCH, 1024, smem1, stream>>>(x, ordidx, tops, boxesW);
    nms_suppress_kernel<<<BATCH, 1024, smem2, stream>>>(x, ordidx, tops, boxesW, out);
}